// ExpertMixer_64639257805147
// MI455X (gfx1250) — compile-verified
//
#include <hip/hip_runtime.h>
#include <stddef.h>

// ---------------------------------------------------------------------------
// MoE combine:  out[tok,:] = sum_{k<2} w[tok,k] * expert_outputs[idx[tok,k], tok, :]
//   B=4 S=2048 H=1024 E=8 K=2, tok = b*S+s (so eo offset = e*TOKENS*H + tok*H).
// 33.6 MFLOP vs ~96 MiB traffic -> AI ~ 0.33 FLOP/B; roofline ~4.3 us @23.3TB/s.
// No shared-operand GEMM structure (gathered operand varies per output row), so
// WMMA is inapplicable; optimize pure data movement: coalesced b128, NT cache
// policy (expert_outputs = 256 MiB > 192 MB L2, zero reuse), and the gfx1250
// async global->LDS DMA path (ASYNCcnt-tracked) to double-buffer the gather.
// ---------------------------------------------------------------------------

#define BB 4
#define SS 2048
#define HH 1024
#define EE 8
#define KK 2
#define TOKENS (BB * SS)      // 8192
#define TPB 4                 // tokens per workgroup
#define THREADS 256           // 8 wave32 waves; thread t owns float4 column t
#define NBLOCKS (TOKENS / TPB)

typedef float f32x4 __attribute__((ext_vector_type(4)));
typedef int   v4i   __attribute__((vector_size(16)));   // v4i32, the builtin's pointee

#if defined(__HIP_DEVICE_COMPILE__) && defined(__gfx1250__)
#if __has_builtin(__builtin_amdgcn_global_load_async_to_lds_b128) && \
    __has_builtin(__builtin_amdgcn_s_wait_asynccnt)
#define HAVE_ASYNC_LDS 1
#endif
#endif
#ifndef HAVE_ASYNC_LDS
#define HAVE_ASYNC_LDS 0
#endif

__global__ __launch_bounds__(THREADS) void moe_combine_kernel(
    const int* __restrict__ eidx,    // [TOKENS, K] int32
    const float* __restrict__ ew,    // [TOKENS, K] f32 (pre-normalized)
    const float* __restrict__ eo,    // [E, TOKENS, H] f32
    float* __restrict__ out)         // [TOKENS, H] f32
{
  const int t = threadIdx.x;                 // float4 column within the H row
  const int tok0 = blockIdx.x * TPB;

#if HAVE_ASYNC_LDS
  // Double-buffered async gather. Stage s holds, per thread, the 16B chunk of
  // each of the two gathered rows. Each thread reads back exactly the bytes
  // its lane DMA'd, so per-wave s_wait_asynccnt is the only sync required
  // (no cross-wave sharing, no workgroup barrier).
  __shared__ f32x4 lbuf[2][2][THREADS];      // 16 KiB of the WGP's 320 KB LDS

  auto issue = [&](int stage, int tok) {
    const int i0 = eidx[tok * KK + 0];
    const int i1 = eidx[tok * KK + 1];
    const float* g0 = eo + ((size_t)i0 * TOKENS + tok) * HH + t * 4;
    const float* g1 = eo + ((size_t)i1 * TOKENS + tok) * HH + t * 4;
    __builtin_amdgcn_global_load_async_to_lds_b128(
        (__attribute__((address_space(1))) v4i*)g0,
        (__attribute__((address_space(3))) v4i*)&lbuf[stage][0][t],
        /*imm offset*/ 0, /*cpol*/ 0);
    __builtin_amdgcn_global_load_async_to_lds_b128(
        (__attribute__((address_space(1))) v4i*)g1,
        (__attribute__((address_space(3))) v4i*)&lbuf[stage][1][t],
        /*imm offset*/ 0, /*cpol*/ 0);
  };

  issue(0, tok0);                            // prologue: stage 0 in flight
#pragma unroll
  for (int i = 0; i < TPB; ++i) {
    const int tok = tok0 + i;
    if (i + 1 < TPB) {
      // Ensure our ds_reads of the stage we are about to overwrite
      // (2 iterations back) have drained before the new DMA can land,
      // then keep 2 tokens (4 async ops) in flight.
      asm volatile("s_wait_dscnt 0" ::: "memory");
      issue((i + 1) & 1, tok + 1);
      __builtin_amdgcn_s_wait_asynccnt(2);   // current stage's 2 DMAs done
    } else {
      __builtin_amdgcn_s_wait_asynccnt(0);
    }
    const float w0 = ew[tok * KK + 0];
    const float w1 = ew[tok * KK + 1];
    f32x4 a = lbuf[i & 1][0][t];
    f32x4 b = lbuf[i & 1][1][t];
    f32x4 o = a * w0 + b * w1;
    __builtin_nontemporal_store(o, (f32x4*)(out + (size_t)tok * HH) + t);
  }
#else
  // Direct path: 2x b128 NT gather loads + fma + b128 NT store per token.
  // Unrolled over TPB tokens for memory-level parallelism within one thread.
#pragma unroll
  for (int i = 0; i < TPB; ++i) {
    const int tok = tok0 + i;
    const int i0 = eidx[tok * KK + 0];
    const int i1 = eidx[tok * KK + 1];
    const float w0 = ew[tok * KK + 0];
    const float w1 = ew[tok * KK + 1];
    const f32x4* r0 = (const f32x4*)(eo + ((size_t)i0 * TOKENS + tok) * HH) + t;
    const f32x4* r1 = (const f32x4*)(eo + ((size_t)i1 * TOKENS + tok) * HH) + t;
    f32x4 a = __builtin_nontemporal_load(r0);
    f32x4 b = __builtin_nontemporal_load(r1);
    f32x4 o = a * w0 + b * w1;
    __builtin_nontemporal_store(o, (f32x4*)(out + (size_t)tok * HH) + t);
  }
#endif
}

extern "C" void kernel_launch(void* const* d_in, const int* in_sizes, int n_in,
                              void* d_out, int out_size, void* d_ws, size_t ws_size,
                              hipStream_t stream) {
  // setup_inputs() order: [0]=hidden_states (unused by reference),
  // [1]=expert_indices, [2]=expert_weights, [3]=expert_outputs.
  const int*   eidx = (const int*)  d_in[1];
  const float* ew   = (const float*)d_in[2];
  const float* eo   = (const float*)d_in[3];
  float*       outp = (float*)d_out;

  moe_combine_kernel<<<NBLOCKS, THREADS, 0, stream>>>(eidx, ew, eo, outp);

  (void)in_sizes; (void)n_in; (void)out_size; (void)d_ws; (void)ws_size;
}